// ClasswiseECELoss_20916490731796
// MI455X (gfx1250) — compile-verified
//
#include <hip/hip_runtime.h>
#include <hip/hip_bf16.h>
#include <math.h>

// ---------------------------------------------------------------------------
// Classwise ECE for MI455X (gfx1250).
//   result = (1/(C*N)) * sum_{c,b} | conf_sum[c,b] - correct_sum[c,b] |
// (counts cancel out of the reference formula.)
//
// Pipeline (all on `stream`):
//   1) zero_kernel      : zero 2*C*15 floats of workspace
//   2) hist_kernel      : TDM-tiled streaming of conf, per-thread(=per-class)
//                         LDS bin sums, global f32 atomics to g_sum
//   3) correct_kernel   : one lookup per sample -> atomics to g_cor
//   4) finalize_kernel  : single-block reduction -> d_out[0]
// ---------------------------------------------------------------------------

#define TPB     256     // threads per block (8 waves of 32)
#define T_ROWS  12      // rows per TDM tile
#define NBINS   15
#define HSTRIDE 17      // odd stride -> conflict-free lane bases in LDS

typedef unsigned int v4u __attribute__((ext_vector_type(4)));
typedef int          v8i __attribute__((ext_vector_type(8)));
typedef int          v4i __attribute__((ext_vector_type(4)));

// Build a 2D Tensor-DMA descriptor (D#) and issue TENSOR_LOAD_TO_LDS.
// Group0 (CDNA5 ISA): count=1 | lds_addr | global_addr | type=2
// Group1 (§8.4): data_size=4B, tensor_dim0/1 (OOB zero-fill), tile_dim0/1,
//                tensor_dim0_stride (elements). Groups 2/3 zero (<=2D tensor).
// Toolchain here uses the 6-arg builtin form (clang-23 / therock-10.0).
__device__ __forceinline__ void tdm_load_tile_2d(unsigned lds_byte_addr,
                                                 const float* gptr,
                                                 unsigned tensor_dim0,   // elems
                                                 unsigned tensor_dim1,   // rows
                                                 unsigned stride0,       // elems
                                                 unsigned tile0,         // elems
                                                 unsigned tile1)         // rows
{
  unsigned long long ga = (unsigned long long)(uintptr_t)gptr;
  v4u g0;
  g0.x = 1u;                                   // count=1 (valid), no gather
  g0.y = lds_byte_addr;                        // LDS destination (bytes)
  g0.z = (unsigned)ga;                         // global_addr[31:0]
  g0.w = (unsigned)(ga >> 32) | (2u << 30);    // global_addr[56:32] | type=2
  v8i g1;
  g1[0] = (int)(2u << 16);                                         // data_size=4B
  g1[1] = (int)((tensor_dim0 & 0xFFFFu) << 16);                    // dim0[15:0]
  g1[2] = (int)((tensor_dim0 >> 16) | ((tensor_dim1 & 0xFFFFu) << 16));
  g1[3] = (int)((tensor_dim1 >> 16) | ((tile0 & 0xFFFFu) << 16));  // tile_dim0
  g1[4] = (int)(tile1 & 0xFFFFu);                                  // tile_dim1 (tile_dim2=0)
  g1[5] = (int)stride0;                                            // dim0_stride[31:0]
  g1[6] = 0;
  g1[7] = 0;
  v4i gz4 = (v4i){0, 0, 0, 0};
  v8i gz8 = (v8i){0, 0, 0, 0, 0, 0, 0, 0};
  __builtin_amdgcn_tensor_load_to_lds(g0, g1, gz4, gz4, gz8, 0);
}

__device__ __forceinline__ int bin_of(float x) {
  int b = (int)ceilf(x * (float)NBINS) - 1;   // conf in (b/15,(b+1)/15] -> b
  return (b > NBINS - 1) ? (NBINS - 1) : b;   // b < 0  => invalid (conf<=0)
}

__global__ void zero_kernel(float* __restrict__ p, int n) {
  int i = blockIdx.x * TPB + threadIdx.x;
  if (i < n) p[i] = 0.0f;
}

// grid = (RB row-blocks, CB class-blocks of 256). Thread t owns class
// c = blockIdx.y*256 + t. TDM streams [T_ROWS x 256] tiles (row stride C),
// double-buffered; OOB rows/classes zero-fill -> conf<=0 -> skipped.
__global__ void __launch_bounds__(TPB)
hist_kernel(const float* __restrict__ conf, float* __restrict__ g_sum,
            int N, int C, int rows_per_block)   // rows_per_block % T_ROWS == 0
{
  __shared__ float s_hist[TPB * HSTRIDE];
  __shared__ float s_tile[2][T_ROWS * TPB];

  const int t    = threadIdx.x;
  const int c0   = blockIdx.y * TPB;
  const int row0 = blockIdx.x * rows_per_block;
  const int rows = min(rows_per_block, N - row0);
  const int ntiles = (rows + T_ROWS - 1) / T_ROWS;
  const bool wave0 = (threadIdx.x >> 5) == 0;

  #pragma unroll
  for (int i = 0; i < HSTRIDE; ++i) s_hist[t * HSTRIDE + i] = 0.0f;
  __syncthreads();

  const unsigned tdim0 = (unsigned)(C - c0);

  if (wave0) {
    const int r = row0;
    tdm_load_tile_2d((unsigned)(uintptr_t)&s_tile[0][0],
                     conf + (size_t)r * C + c0,
                     tdim0, (unsigned)(N - r), (unsigned)C, TPB, T_ROWS);
    __builtin_amdgcn_s_wait_tensorcnt(0);
  }
  __syncthreads();

  for (int k = 0; k < ntiles; ++k) {
    const int cur = k & 1;
    if (wave0 && (k + 1) < ntiles) {
      const int r = row0 + (k + 1) * T_ROWS;
      tdm_load_tile_2d((unsigned)(uintptr_t)&s_tile[cur ^ 1][0],
                       conf + (size_t)r * C + c0,
                       tdim0, (unsigned)(N - r), (unsigned)C, TPB, T_ROWS);
    }

    const float* buf = &s_tile[cur][0];
    #pragma unroll
    for (int r = 0; r < T_ROWS; ++r) {
      float x = buf[r * TPB + t];
      int b = bin_of(x);
      if (b >= 0) atomicAdd(&s_hist[t * HSTRIDE + b], x);  // ds_add_f32, no contention
    }

    if (wave0 && (k + 1) < ntiles) __builtin_amdgcn_s_wait_tensorcnt(0);
    __syncthreads();
  }

  const int c = c0 + t;
  if (c < C) {
    #pragma unroll
    for (int b = 0; b < NBINS; ++b) {
      float v = s_hist[t * HSTRIDE + b];
      if (v != 0.0f) atomicAdd(&g_sum[c * NBINS + b], v);  // global_atomic_add_f32
    }
  }
}

__global__ void correct_kernel(const float* __restrict__ conf,
                               const int* __restrict__ labels,
                               float* __restrict__ g_cor, int N, int C)
{
  int n = blockIdx.x * TPB + threadIdx.x;
  if (n < N) {
    int c = labels[n];
    float x = conf[(size_t)n * C + c];
    int b = bin_of(x);
    if (b >= 0) atomicAdd(&g_cor[c * NBINS + b], 1.0f);
  }
}

__global__ void finalize_kernel(const float* __restrict__ g_sum,
                                const float* __restrict__ g_cor,
                                float* __restrict__ out, int total, float scale)
{
  __shared__ float red[TPB];
  float acc = 0.0f;
  for (int i = threadIdx.x; i < total; i += TPB)
    acc += fabsf(g_sum[i] - g_cor[i]);
  red[threadIdx.x] = acc;
  __syncthreads();
  for (int s = TPB / 2; s > 0; s >>= 1) {
    if (threadIdx.x < s) red[threadIdx.x] += red[threadIdx.x + s];
    __syncthreads();
  }
  if (threadIdx.x == 0) out[0] = red[0] * scale;
}

extern "C" void kernel_launch(void* const* d_in, const int* in_sizes, int n_in,
                              void* d_out, int out_size, void* d_ws, size_t ws_size,
                              hipStream_t stream) {
  const float* conf   = (const float*)d_in[0];
  const int*   labels = (const int*)d_in[1];
  float*       out    = (float*)d_out;

  const int N = in_sizes[1];          // labels: [N]
  const int C = in_sizes[0] / N;      // conf:   [N, C]
  const int nseg = C * NBINS;

  float* g_sum = (float*)d_ws;        // [C*15]
  float* g_cor = g_sum + nseg;        // [C*15]

  // 1) zero workspace accumulators
  {
    int zn = 2 * nseg;
    zero_kernel<<<dim3((zn + TPB - 1) / TPB), dim3(TPB), 0, stream>>>(g_sum, zn);
  }

  // 2) conf_sum histogram (TDM-tiled)
  {
    const int rpb = 65 * T_ROWS;      // 780, multiple of T_ROWS
    const int RB  = (N + rpb - 1) / rpb;
    const int CB  = (C + TPB - 1) / TPB;
    hist_kernel<<<dim3(RB, CB), dim3(TPB), 0, stream>>>(conf, g_sum, N, C, rpb);
  }

  // 3) correct_sum
  correct_kernel<<<dim3((N + TPB - 1) / TPB), dim3(TPB), 0, stream>>>(conf, labels, g_cor, N, C);

  // 4) finalize scalar
  finalize_kernel<<<dim3(1), dim3(TPB), 0, stream>>>(
      g_sum, g_cor, out, nseg, 1.0f / ((float)C * (float)N));
}